// EQGATGNN_33363305956112
// MI455X (gfx1250) — compile-verified
//
#include <hip/hip_runtime.h>
#include <hip/hip_bf16.h>

#define SI 128
#define VI 32
#define NR 32
#define NNODES 10000
#define NEDGES 160000
#define XK (2*SI+NR)     // 288
#define CUTOFF_F 5.0f
#define PI_F 3.14159265358979323846f

typedef __attribute__((ext_vector_type(2))) float v2f;
typedef __attribute__((ext_vector_type(8))) float v8f;

// D = A(16x4,f32) * B(4x16,f32) + C(16x16,f32)  -- full fp32 WMMA
__device__ __forceinline__ v8f wmma4(v8f acc, v2f a, v2f b) {
  return __builtin_amdgcn_wmma_f32_16x16x4_f32(false, a, false, b, (short)0, acc,
                                               false, false);
}
__device__ __forceinline__ float siluf(float x) { return x / (1.f + __expf(-x)); }
__device__ __forceinline__ void atomicMaxF(float* a, float v) {
  if (v >= 0.f) atomicMax((int*)a, __float_as_int(v));
  else          atomicMin((unsigned int*)a, __float_as_uint(v));
}

// 1x1 K-segment (small GEMMs)
template <int K>
__device__ __forceinline__ v8f core(const float* __restrict__ Ar,
                                    const float* __restrict__ Br,
                                    int koff, v8f acc) {
#pragma unroll
  for (int k = 0; k < K; k += 4) {
    v2f a = *(const v2f*)(Ar + k + koff);
    v2f b = *(const v2f*)(Br + k + koff);
    acc = wmma4(acc, a, b);
  }
  return acc;
}

// 2x2 K-segment: 4 loads feed 4 WMMAs (register-level A/B reuse)
template <int K>
__device__ __forceinline__ void core2(const float* __restrict__ A0,
                                      const float* __restrict__ A1,
                                      const float* __restrict__ B0,
                                      const float* __restrict__ B1, int koff,
                                      v8f& c00, v8f& c01, v8f& c10, v8f& c11) {
#pragma unroll
  for (int k = 0; k < K; k += 4) {
    v2f a0 = *(const v2f*)(A0 + k + koff);
    v2f a1 = *(const v2f*)(A1 + k + koff);
    v2f b0 = *(const v2f*)(B0 + k + koff);
    v2f b1 = *(const v2f*)(B1 + k + koff);
    c00 = wmma4(c00, a0, b0);
    c01 = wmma4(c01, a0, b1);
    c10 = wmma4(c10, a1, b0);
    c11 = wmma4(c11, a1, b1);
  }
}

// ------- 1x1 generic dense GEMM: C = act(A[MxK] @ BT[NxK]^T + bias) ---------
template <int K>
__global__ void k_gemm_t(const float* __restrict__ A, const float* __restrict__ BT,
                         const float* __restrict__ bias, float* __restrict__ C,
                         int N, int act) {
  int lane = threadIdx.x & 31;
  int half = lane >> 4, l = lane & 15;
  int row0 = blockIdx.y << 4, col0 = blockIdx.x << 4;
  int col = col0 + l;
  int colc = (col < N) ? col : (N - 1);
  int koff = half << 1;
  const float* Ar = A + (size_t)(row0 + l) * K;
  const float* Br = BT + (size_t)colc * K;
  v8f acc = {};
  acc = core<K>(Ar, Br, koff, acc);
  if (col >= N) return;
  float bv = bias ? bias[col] : 0.f;
#pragma unroll
  for (int r = 0; r < 8; ++r) {
    int row = row0 + r + (half << 3);
    float x = acc[r] + bv;
    if (act) x = siluf(x);
    C[(size_t)row * N + col] = x;
  }
}

// ------- 2x2 generic dense GEMM (32x32 out per wave), row/col guarded -------
template <int K>
__global__ void k_gemm2_t(const float* __restrict__ A, const float* __restrict__ BT,
                          const float* __restrict__ bias, float* __restrict__ C,
                          int M, int N, int act) {
  int lane = threadIdx.x & 31;
  int half = lane >> 4, l = lane & 15;
  int row0 = blockIdx.y << 5, col0 = blockIdx.x << 5;
  int koff = half << 1;
  int r0 = row0 + l, r1 = row0 + 16 + l;
  int r0c = (r0 < M) ? r0 : (M - 1);
  int r1c = (r1 < M) ? r1 : (M - 1);
  int c0 = col0 + l, c1 = col0 + 16 + l;
  int c0c = (c0 < N) ? c0 : (N - 1);
  int c1c = (c1 < N) ? c1 : (N - 1);
  const float* A0 = A + (size_t)r0c * K;
  const float* A1 = A + (size_t)r1c * K;
  const float* B0 = BT + (size_t)c0c * K;
  const float* B1 = BT + (size_t)c1c * K;
  v8f c00 = {}, c01 = {}, c10 = {}, c11 = {};
  core2<K>(A0, A1, B0, B1, koff, c00, c01, c10, c11);
  float bv0 = (bias && c0 < N) ? bias[c0] : 0.f;
  float bv1 = (bias && c1 < N) ? bias[c1] : 0.f;
#pragma unroll
  for (int r = 0; r < 8; ++r) {
    int rowA = row0 + r + (half << 3);
    int rowB = rowA + 16;
    float x;
    if (rowA < M && c0 < N) { x = c00[r] + bv0; if (act) x = siluf(x); C[(size_t)rowA * N + c0] = x; }
    if (rowA < M && c1 < N) { x = c01[r] + bv1; if (act) x = siluf(x); C[(size_t)rowA * N + c1] = x; }
    if (rowB < M && c0 < N) { x = c10[r] + bv0; if (act) x = siluf(x); C[(size_t)rowB * N + c0] = x; }
    if (rowB < M && c1 < N) { x = c11[r] + bv1; if (act) x = siluf(x); C[(size_t)rowB * N + c1] = x; }
  }
}

// --------- edge MLP1: H = silu([s[dst] | s[src] | de] @ We1 + be1), N=128 ---
// WT is We1 transposed: [128][288]; 2x2 tiles, three branch-free K segments.
__global__ void k_gemm_edge1(const float* __restrict__ s, const float* __restrict__ de,
                             const int* __restrict__ srcI, const int* __restrict__ dstI,
                             const float* __restrict__ WT, const float* __restrict__ bias,
                             float* __restrict__ H) {
  int lane = threadIdx.x & 31;
  int half = lane >> 4, l = lane & 15;
  int row0 = blockIdx.y << 5, col0 = blockIdx.x << 5;
  int koff = half << 1;
  int e0 = row0 + l, e1 = row0 + 16 + l;
  const float* sd0 = s + (size_t)dstI[e0] * SI;
  const float* ss0 = s + (size_t)srcI[e0] * SI;
  const float* dr0 = de + (size_t)e0 * NR;
  const float* sd1 = s + (size_t)dstI[e1] * SI;
  const float* ss1 = s + (size_t)srcI[e1] * SI;
  const float* dr1 = de + (size_t)e1 * NR;
  int c0 = col0 + l, c1 = col0 + 16 + l;
  const float* B0 = WT + (size_t)c0 * XK;
  const float* B1 = WT + (size_t)c1 * XK;
  v8f c00 = {}, c01 = {}, c10 = {}, c11 = {};
  core2<SI>(sd0, sd1, B0, B1, koff, c00, c01, c10, c11);
  core2<SI>(ss0, ss1, B0 + SI, B1 + SI, koff, c00, c01, c10, c11);
  core2<NR>(dr0, dr1, B0 + 2 * SI, B1 + 2 * SI, koff, c00, c01, c10, c11);
  float bv0 = bias[c0], bv1 = bias[c1];
#pragma unroll
  for (int r = 0; r < 8; ++r) {
    int rowA = row0 + r + (half << 3);
    int rowB = rowA + 16;
    H[(size_t)rowA * SI + c0] = siluf(c00[r] + bv0);
    H[(size_t)rowA * SI + c1] = siluf(c01[r] + bv1);
    H[(size_t)rowB * SI + c0] = siluf(c10[r] + bv0);
    H[(size_t)rowB * SI + c1] = siluf(c11[r] + bv1);
  }
}

// --------- edge MLP2: a = H @ We2 + be2, scatter into fij / wij / gbuf ------
__device__ __forceinline__ void edge2_store(float* __restrict__ fij,
                                            float* __restrict__ wij,
                                            float* __restrict__ gbuf, int AOUT,
                                            int GW, int e, int col, float x) {
  if (col >= AOUT) return;
  if (col < SI)       fij[(size_t)e * SI + col] = x;
  else if (col == SI) wij[e] = x;
  else                gbuf[(size_t)e * GW + (col - SI - 1)] = x;
}
__global__ void k_gemm_edge2(const float* __restrict__ H, const float* __restrict__ WT,
                             const float* __restrict__ bias, float* __restrict__ fij,
                             float* __restrict__ wij, float* __restrict__ gbuf,
                             int AOUT, int GW) {
  int lane = threadIdx.x & 31;
  int half = lane >> 4, l = lane & 15;
  int row0 = blockIdx.y << 5, col0 = blockIdx.x << 5;
  int koff = half << 1;
  const float* H0 = H + (size_t)(row0 + l) * SI;
  const float* H1 = H + (size_t)(row0 + 16 + l) * SI;
  int c0 = col0 + l, c1 = col0 + 16 + l;
  int c0c = (c0 < AOUT) ? c0 : (AOUT - 1);
  int c1c = (c1 < AOUT) ? c1 : (AOUT - 1);
  const float* B0 = WT + (size_t)c0c * SI;
  const float* B1 = WT + (size_t)c1c * SI;
  v8f c00 = {}, c01 = {}, c10 = {}, c11 = {};
  core2<SI>(H0, H1, B0, B1, koff, c00, c01, c10, c11);
  float bv0 = (c0 < AOUT) ? bias[c0] : 0.f;
  float bv1 = (c1 < AOUT) ? bias[c1] : 0.f;
#pragma unroll
  for (int r = 0; r < 8; ++r) {
    int eA = row0 + r + (half << 3);
    int eB = eA + 16;
    edge2_store(fij, wij, gbuf, AOUT, GW, eA, c0, c00[r] + bv0);
    edge2_store(fij, wij, gbuf, AOUT, GW, eA, c1, c01[r] + bv1);
    edge2_store(fij, wij, gbuf, AOUT, GW, eB, c0, c10[r] + bv0);
    edge2_store(fij, wij, gbuf, AOUT, GW, eB, c1, c11[r] + bv1);
  }
}

// --------- node MLP: T = silu([s | vn] @ Wm1 + bm1), K=160, N=128 -----------
__global__ void k_gemm_cat(const float* __restrict__ s, const float* __restrict__ vn,
                           const float* __restrict__ WT, const float* __restrict__ bias,
                           float* __restrict__ T) {
  int lane = threadIdx.x & 31;
  int half = lane >> 4, l = lane & 15;
  int row0 = blockIdx.y << 4, col0 = blockIdx.x << 4;
  int node = row0 + l;
  const float* sr = s + (size_t)node * SI;
  const float* vr = vn + (size_t)node * VI;
  int col = col0 + l;
  int koff = half << 1;
  const float* Br = WT + (size_t)col * (SI + VI);
  v8f acc = {};
  acc = core<SI>(sr, Br, koff, acc);
  acc = core<VI>(vr, Br + SI, koff, acc);
  float bv = bias[col];
#pragma unroll
  for (int r = 0; r < 8; ++r) {
    int row = row0 + r + (half << 3);
    T[(size_t)row * SI + col] = siluf(acc[r] + bv);
  }
}

// ------------------------------- elementwise --------------------------------
__global__ void k_transp(const float* __restrict__ B, float* __restrict__ BT,
                         int K, int N) {
  int t = blockIdx.x * blockDim.x + threadIdx.x;
  if (t >= K * N) return;
  int k = t / N, n = t - k * N;
  BT[(size_t)n * K + k] = B[t];
}
__global__ void k_fill(float* __restrict__ p, float v, int n) {
  int t = blockIdx.x * blockDim.x + threadIdx.x;
  if (t < n) p[t] = v;
}
__global__ void k_copy(float* __restrict__ d, const float* __restrict__ s, int n) {
  int t = blockIdx.x * blockDim.x + threadIdx.x;
  if (t < n) d[t] = s[t];
}
__global__ void k_rbf(const float* __restrict__ d, float* __restrict__ de) {
  int t = blockIdx.x * blockDim.x + threadIdx.x;
  if (t >= NEDGES * NR) return;
  int e = t >> 5, j = t & 31;
  float dd = d[e];
  float dsafe = fmaxf(dd, 1e-6f);
  float s = sqrtf(2.f / CUTOFF_F) * __sinf((float)(j + 1) * (PI_F / CUTOFF_F) * dsafe) / dsafe;
  float dc = 0.5f * (__cosf(PI_F * dd / CUTOFF_F) + 1.f) * (dd < CUTOFF_F ? 1.f : 0.f);
  de[t] = dc * s;
}
__global__ void k_segmax(const float* __restrict__ fij, const int* __restrict__ dstI,
                         float* __restrict__ mmax) {
  int t = blockIdx.x * blockDim.x + threadIdx.x;
  if (t >= NEDGES * SI) return;
  int e = t >> 7, f = t & (SI - 1);
  atomicMaxF(&mmax[(size_t)dstI[e] * SI + f], fij[t]);
}
__global__ void k_segexp(float* __restrict__ fij, const int* __restrict__ dstI,
                         const float* __restrict__ mmax, float* __restrict__ z,
                         float* __restrict__ deg) {
  int t = blockIdx.x * blockDim.x + threadIdx.x;
  if (t >= NEDGES * SI) return;
  int e = t >> 7, f = t & (SI - 1);
  int dn = dstI[e];
  float ev = __expf(fij[t] - mmax[(size_t)dn * SI + f]);
  fij[t] = ev;
  atomicAdd(&z[(size_t)dn * SI + f], ev);
  if (f == 0) atomicAdd(&deg[dn], 1.f);
}
__global__ void k_scatter_s(const float* __restrict__ e_, const float* __restrict__ z,
                            const float* __restrict__ sb, const int* __restrict__ srcI,
                            const int* __restrict__ dstI, float* __restrict__ s_nxt) {
  int t = blockIdx.x * blockDim.x + threadIdx.x;
  if (t >= NEDGES * SI) return;
  int e = t >> 7, f = t & (SI - 1);
  int dn = dstI[e], sn = srcI[e];
  float ns = e_[t] / (z[(size_t)dn * SI + f] + 1e-16f) * sb[(size_t)sn * SI + f];
  atomicAdd(&s_nxt[(size_t)dn * SI + f], ns);
}
__global__ void k_scatter_v(const float* __restrict__ g, const float* __restrict__ wij,
                            const float* __restrict__ r, const float* __restrict__ v,
                            const float* __restrict__ vb, const int* __restrict__ srcI,
                            const int* __restrict__ dstI, float* __restrict__ vacc,
                            int hasv) {
  int t = blockIdx.x * blockDim.x + threadIdx.x;
  if (t >= NEDGES * VI) return;
  int e = t >> 5, i = t & 31;
  int dn = dstI[e], sn = srcI[e];
  float sg = 1.f / (1.f + __expf(-wij[e]));
  float g0 = g[(size_t)e * 96 + i];
  float nv0 = g0 * r[e * 3 + 0], nv1 = g0 * r[e * 3 + 1], nv2 = g0 * r[e * 3 + 2];
  if (hasv) {
    float g1 = g[(size_t)e * 96 + 32 + i];
    float g2 = g[(size_t)e * 96 + 64 + i];
    float vb0 = vb[(size_t)sn * 96 + i], vb1 = vb[(size_t)sn * 96 + 32 + i],
          vb2 = vb[(size_t)sn * 96 + 64 + i];
    float vd0 = v[(size_t)dn * 96 + i], vd1 = v[(size_t)dn * 96 + 32 + i],
          vd2 = v[(size_t)dn * 96 + 64 + i];
    nv0 += g1 * vb0 + g2 * (vd1 * vb2 - vd2 * vb1);
    nv1 += g1 * vb1 + g2 * (vd2 * vb0 - vd0 * vb2);
    nv2 += g1 * vb2 + g2 * (vd0 * vb1 - vd1 * vb0);
  }
  atomicAdd(&vacc[(size_t)dn * 96 + i], sg * nv0);
  atomicAdd(&vacc[(size_t)dn * 96 + 32 + i], sg * nv1);
  atomicAdd(&vacc[(size_t)dn * 96 + 64 + i], sg * nv2);
}
__global__ void k_v_update(float* __restrict__ v_nxt, const float* __restrict__ v_cur,
                           const float* __restrict__ vacc, const float* __restrict__ deg) {
  int t = blockIdx.x * blockDim.x + threadIdx.x;
  if (t >= NNODES * 96) return;
  int n = t / 96;
  v_nxt[t] = v_cur[t] + vacc[t] / fmaxf(deg[n], 1.f);
}
__global__ void k_vnorm(const float* __restrict__ vv, float* __restrict__ vn) {
  int t = blockIdx.x * blockDim.x + threadIdx.x;
  if (t >= NNODES * VI) return;
  int n = t >> 5, j = t & 31;
  float a0 = vv[(size_t)n * 192 + j];
  float a1 = vv[(size_t)n * 192 + 64 + j];
  float a2 = vv[(size_t)n * 192 + 128 + j];
  vn[t] = sqrtf(fmaxf(a0 * a0 + a1 * a1 + a2 * a2, 1e-6f));
}
__global__ void k_final_update(float* __restrict__ s, float* __restrict__ v,
                               const float* __restrict__ m, const float* __restrict__ vv) {
  int t = blockIdx.x * blockDim.x + threadIdx.x;
  const int W = SI + 96;
  if (t >= NNODES * W) return;
  int node = t / W;
  int r = t - node * W;
  if (r < SI) {
    s[(size_t)node * SI + r] += m[(size_t)node * (VI + SI) + VI + r];
  } else {
    int q = r - SI;
    int c = q >> 5, j = q & 31;
    v[(size_t)node * 96 + q] += m[(size_t)node * (VI + SI) + j] *
                                vv[(size_t)node * 192 + c * 64 + VI + j];
  }
}

// --------------------------------- host -------------------------------------
extern "C" void kernel_launch(void* const* d_in, const int* in_sizes, int n_in,
                              void* d_out, int out_size, void* d_ws, size_t ws_size,
                              hipStream_t stream) {
  const float* s_in = (const float*)d_in[0];
  const float* v_in = (const float*)d_in[1];
  const int*   eidx = (const int*)d_in[2];
  const float* dvec = (const float*)d_in[3];
  const float* rvec = (const float*)d_in[4];
  const int* srcI = eidx;
  const int* dstI = eidx + NEDGES;

  // Param slots: 0=We1 1=be1 2=We2 3=be2 4=Ws 5=bs 6=Wv0 7=Wm1 8=bm1 9=Wm2 10=bm2 11=Wv
  static const int mapIns[12]    = {0, 1, 2, 3, 4, 5, 6, 7, 8, 9, 10, 11};
  static const int mapSort12[12] = {0, 7, 1, 8, 4, 11, 6, 2, 9, 3, 10, 5};
  static const int mapSort11[12] = {0, 6, 1, 7, 4, 10, 5, 2, 8, 3, 9, -1};
  const float* P[5][12];
  int idx = 6;
  for (int layer = 0; layer < 5; ++layer) {
    bool hasv = layer > 0;
    int cnt = hasv ? 12 : 11;
    bool ins = (idx + 1 < n_in) && (in_sizes[idx + 1] == SI);  // be1 at +1 => insertion
    for (int slot = 0; slot < 12; ++slot) {
      int off;
      if (ins) off = (slot == 11 && !hasv) ? -1 : mapIns[slot];
      else     off = hasv ? mapSort12[slot] : mapSort11[slot];
      P[layer][slot] = (off >= 0) ? (const float*)d_in[idx + off] : nullptr;
    }
    idx += cnt;
  }

  // workspace carve-up
  float* w = (float*)d_ws;
  auto take = [&](size_t nfl) { float* p = w; w += nfl; return p; };
  float* sbuf0 = take((size_t)NNODES * SI);
  float* sbuf1 = take((size_t)NNODES * SI);
  float* vbuf0 = take((size_t)NNODES * 96);
  float* vbuf1 = take((size_t)NNODES * 96);
  float* sb    = take((size_t)NNODES * SI);
  float* vbb   = take((size_t)NNODES * 96);
  float* debuf = take((size_t)NEDGES * NR);
  float* hbuf  = take((size_t)NEDGES * SI);
  float* fij   = take((size_t)NEDGES * SI);
  float* wijb  = take((size_t)NEDGES);
  float* gbuf  = take((size_t)NEDGES * 96);
  float* mmax  = take((size_t)NNODES * SI);
  float* zbuf  = take((size_t)NNODES * SI);
  float* degb  = take((size_t)NNODES);
  float* vacc  = take((size_t)NNODES * 96);
  float* vvb   = take((size_t)NNODES * 192);
  float* vnb   = take((size_t)NNODES * VI);
  float* tbuf  = take((size_t)NNODES * SI);
  float* mbuf  = take((size_t)NNODES * (VI + SI));
  // transposed weights
  float* We1T  = take((size_t)XK * SI);        // [128][288]
  float* We2T  = take((size_t)SI * 240);       // [AOUT][128], AOUT<=225
  float* WsT   = take((size_t)SI * SI);
  float* WvT   = take((size_t)VI * VI);
  float* Wv0T  = take((size_t)VI * 2 * VI);
  float* Wm1T  = take((size_t)(SI + VI) * SI);
  float* Wm2T  = take((size_t)SI * (VI + SI));

  auto ew = [&](int n) { return dim3((unsigned)((n + 255) / 256)); };
  auto transp = [&](const float* B, float* BT, int K, int N) {
    k_transp<<<ew(K * N), 256, 0, stream>>>(B, BT, K, N);
  };

  float* s_cur = sbuf0; float* s_nxt = sbuf1;
  float* v_cur = vbuf0; float* v_nxt = vbuf1;
  k_copy<<<ew(NNODES * SI), 256, 0, stream>>>(s_cur, s_in, NNODES * SI);
  k_copy<<<ew(NNODES * 96), 256, 0, stream>>>(v_cur, v_in, NNODES * 96);

  for (int layer = 0; layer < 5; ++layer) {
    bool hasv = layer > 0;
    const float* const* p = P[layer];
    int v_mul = hasv ? 3 : 1;
    int AOUT = SI + 1 + v_mul * VI;  // 161 or 225

    // transpose layer weights to [N][K]
    transp(p[0], We1T, XK, SI);
    transp(p[2], We2T, SI, AOUT);
    transp(p[4], WsT, SI, SI);
    if (hasv) transp(p[11], WvT, VI, VI);
    transp(p[6], Wv0T, VI, 2 * VI);
    transp(p[7], Wm1T, SI + VI, SI);
    transp(p[9], Wm2T, SI, VI + SI);

    // sb = s@Ws + bs   (2x2 tiles, M=10000 guarded)
    k_gemm2_t<SI><<<dim3(SI / 32, (NNODES + 31) / 32), 32, 0, stream>>>(
        s_cur, WsT, p[5], sb, NNODES, SI, 0);
    // vb = v@Wv
    if (hasv)
      k_gemm_t<VI><<<dim3(VI / 16, 3 * NNODES / 16), 32, 0, stream>>>(
          v_cur, WvT, nullptr, vbb, VI, 0);
    k_rbf<<<ew(NEDGES * NR), 256, 0, stream>>>(dvec, debuf);

    k_gemm_edge1<<<dim3(SI / 32, NEDGES / 32), 32, 0, stream>>>(
        s_cur, debuf, srcI, dstI, We1T, p[1], hbuf);
    k_gemm_edge2<<<dim3((AOUT + 31) / 32, NEDGES / 32), 32, 0, stream>>>(
        hbuf, We2T, p[3], fij, wijb, gbuf, AOUT, 96);

    k_fill<<<ew(NNODES * SI), 256, 0, stream>>>(mmax, -3.0e38f, NNODES * SI);
    k_fill<<<ew(NNODES * SI), 256, 0, stream>>>(zbuf, 0.f, NNODES * SI);
    k_fill<<<ew(NNODES), 256, 0, stream>>>(degb, 0.f, NNODES);
    k_fill<<<ew(NNODES * 96), 256, 0, stream>>>(vacc, 0.f, NNODES * 96);
    k_copy<<<ew(NNODES * SI), 256, 0, stream>>>(s_nxt, s_cur, NNODES * SI);

    k_segmax<<<ew(NEDGES * SI), 256, 0, stream>>>(fij, dstI, mmax);
    k_segexp<<<ew(NEDGES * SI), 256, 0, stream>>>(fij, dstI, mmax, zbuf, degb);
    k_scatter_s<<<ew(NEDGES * SI), 256, 0, stream>>>(fij, zbuf, sb, srcI, dstI, s_nxt);
    k_scatter_v<<<ew(NEDGES * VI), 256, 0, stream>>>(
        gbuf, wijb, rvec, v_cur, hasv ? vbb : v_cur, srcI, dstI, vacc, hasv ? 1 : 0);
    k_v_update<<<ew(NNODES * 96), 256, 0, stream>>>(v_nxt, v_cur, vacc, degb);

    // vv = v@Wv0
    k_gemm_t<VI><<<dim3(2 * VI / 16, 3 * NNODES / 16), 32, 0, stream>>>(
        v_nxt, Wv0T, nullptr, vvb, 2 * VI, 0);
    k_vnorm<<<ew(NNODES * VI), 256, 0, stream>>>(vvb, vnb);
    k_gemm_cat<<<dim3(SI / 16, NNODES / 16), 32, 0, stream>>>(
        s_nxt, vnb, Wm1T, p[8], tbuf);
    // m = t@Wm2 + bm2  (2x2 tiles, N=160, M=10000 guarded)
    k_gemm2_t<SI><<<dim3((VI + SI) / 32, (NNODES + 31) / 32), 32, 0, stream>>>(
        tbuf, Wm2T, p[10], mbuf, NNODES, VI + SI, 0);
    k_final_update<<<ew(NNODES * (SI + 96)), 256, 0, stream>>>(s_nxt, v_nxt, mbuf, vvb);

    float* ts = s_cur; s_cur = s_nxt; s_nxt = ts;
    float* tv = v_cur; v_cur = v_nxt; v_nxt = tv;
  }

  float* out = (float*)d_out;
  k_copy<<<ew(NNODES * SI), 256, 0, stream>>>(out, s_cur, NNODES * SI);
  k_copy<<<ew(NNODES * 96), 256, 0, stream>>>(out + (size_t)NNODES * SI, v_cur, NNODES * 96);
}